// GATLayer_82325933129885
// MI455X (gfx1250) — compile-verified
//
#include <hip/hip_runtime.h>
#include <math.h>

#define N_NODES 50000
#define N_EDGES 500000
#define IN_DIM 256
#define OUT_DIM 128
#define NEG_SLOPE 0.01f

typedef float v2f __attribute__((ext_vector_type(2)));
typedef float v8f __attribute__((ext_vector_type(8)));
typedef int   v4i __attribute__((ext_vector_type(4)));

// gfx1250 async global->LDS path (ASYNCcnt), gated so the file always compiles.
#if defined(__has_builtin)
#if __has_builtin(__builtin_amdgcn_global_load_async_to_lds_b128) && \
    __has_builtin(__builtin_amdgcn_s_wait_asynccnt)
#define USE_ASYNC_LDS 1
#endif
#endif
#ifndef USE_ASYNC_LDS
#define USE_ASYNC_LDS 0
#endif

#if USE_ASYNC_LDS
// Builtin signature: (v4i AS1*, v4i AS3*, imm int offset, imm int cpol)
typedef __attribute__((address_space(1))) v4i gv4i;  // global
typedef __attribute__((address_space(3))) v4i lv4i;  // LDS
#endif

// ---------------------------------------------------------------------------
// Kernel 1: z = h @ fc_w^T via V_WMMA_F32_16X16X4_F32 (fp32 tensor path).
// Block = 256 threads (8 wave32), tile = 128 rows x 128 cols, K chunked by 32
// through LDS (padded stride 36 -> conflict-free fragment reads).
// Async path: double-buffered GLOBAL_LOAD_ASYNC_TO_LDS_B128, next chunk's copy
// overlapped with current chunk's WMMAs via s_wait_asynccnt(8).
// ---------------------------------------------------------------------------
#define KCH 32
#define NCHUNK (IN_DIM / KCH)  // 8
#define LDSS 36                // 36*4 = 144 B, multiple of 16 -> b128 aligned

__launch_bounds__(256)
__global__ void gat_gemm_z(const float* __restrict__ h,
                           const float* __restrict__ fcw,
                           float* __restrict__ z) {
#if USE_ASYNC_LDS
  __shared__ float hs[2][128 * LDSS];
  __shared__ float wls[2][128 * LDSS];
#else
  __shared__ float hs[1][128 * LDSS];
  __shared__ float wls[1][128 * LDSS];
#endif
  const int t    = threadIdx.x;
  const int lane = t & 31;
  const int wave = t >> 5;
  const int row0 = blockIdx.x * 128;

  const v8f zero = {0.f, 0.f, 0.f, 0.f, 0.f, 0.f, 0.f, 0.f};
  v8f acc[8];
#pragma unroll
  for (int i = 0; i < 8; ++i) acc[i] = zero;

  // Fragment coordinates.
  const int m_l   = (wave << 4) + (lane & 15);  // A row within 128-row tile
  const int kbase = (lane >> 4) << 1;           // 0 (lanes 0-15) / 2 (lanes 16-31)

#if USE_ASYNC_LDS
  // Issue one K-chunk's tiles (128x32 floats each) as 8 async b128 per thread.
  auto issue_chunk = [&](int c, int buf) {
#pragma unroll
    for (int i = 0; i < 4; ++i) {
      const int linear = i * 256 + t;   // float4 slot
      const int r  = linear >> 3;       // 8 float4 per row
      const int c4 = (linear & 7) << 2; // float column
      const int k0 = c * KCH;
      // Clamp OOB tail rows (values unused; output stores are guarded).
      const int hr = (row0 + r < N_NODES) ? (row0 + r) : (N_NODES - 1);
      const float* gh = h + (size_t)hr * IN_DIM + k0 + c4;
      __builtin_amdgcn_global_load_async_to_lds_b128(
          (gv4i*)gh, (lv4i*)&hs[buf][r * LDSS + c4], 0, 0);
      const float* gw = fcw + (size_t)r * IN_DIM + k0 + c4;
      __builtin_amdgcn_global_load_async_to_lds_b128(
          (gv4i*)gw, (lv4i*)&wls[buf][r * LDSS + c4], 0, 0);
    }
  };

  issue_chunk(0, 0);
  for (int c = 0; c < NCHUNK; ++c) {
    const int cur = c & 1;
    if (c + 1 < NCHUNK) {
      issue_chunk(c + 1, cur ^ 1);             // prefetch overlaps compute
      __builtin_amdgcn_s_wait_asynccnt(8);     // in-order: chunk c complete
    } else {
      __builtin_amdgcn_s_wait_asynccnt(0);
    }
    __syncthreads();
#pragma unroll
    for (int kk = 0; kk < KCH; kk += 4) {
      const v2f a = *(const v2f*)(&hs[cur][m_l * LDSS + kk + kbase]);
#pragma unroll
      for (int nt = 0; nt < 8; ++nt) {
        const int n_l = (nt << 4) + (lane & 15);
        const v2f b = *(const v2f*)(&wls[cur][n_l * LDSS + kk + kbase]);
        acc[nt] = __builtin_amdgcn_wmma_f32_16x16x4_f32(
            false, a, false, b, (short)0, acc[nt], false, false);
      }
    }
    __syncthreads();  // protect buffer before next prefetch overwrites it
  }
#else
  // Fallback: global -> VGPR -> ds_store staging, single buffer.
  for (int k0 = 0; k0 < IN_DIM; k0 += KCH) {
#pragma unroll
    for (int i = 0; i < 4; ++i) {
      const int linear = i * 256 + t;
      const int r  = linear >> 3;
      const int c4 = (linear & 7) << 2;
      float4 hv = make_float4(0.f, 0.f, 0.f, 0.f);
      if (row0 + r < N_NODES)
        hv = *(const float4*)(h + (size_t)(row0 + r) * IN_DIM + k0 + c4);
      *(float4*)(&hs[0][r * LDSS + c4]) = hv;
      const float4 wv = *(const float4*)(fcw + (size_t)r * IN_DIM + k0 + c4);
      *(float4*)(&wls[0][r * LDSS + c4]) = wv;
    }
    __syncthreads();
#pragma unroll
    for (int kk = 0; kk < KCH; kk += 4) {
      const v2f a = *(const v2f*)(&hs[0][m_l * LDSS + kk + kbase]);
#pragma unroll
      for (int nt = 0; nt < 8; ++nt) {
        const int n_l = (nt << 4) + (lane & 15);
        const v2f b = *(const v2f*)(&wls[0][n_l * LDSS + kk + kbase]);
        acc[nt] = __builtin_amdgcn_wmma_f32_16x16x4_f32(
            false, a, false, b, (short)0, acc[nt], false, false);
      }
    }
    __syncthreads();
  }
#endif

  // C/D layout: VGPR r -> M = r (lanes 0-15) / M = 8+r (lanes 16-31); N = lane&15.
  const int mbase = row0 + (wave << 4) + ((lane >> 4) << 3);
  const int ncol  = lane & 15;
#pragma unroll
  for (int nt = 0; nt < 8; ++nt) {
#pragma unroll
    for (int r = 0; r < 8; ++r) {
      const int row = mbase + r;
      if (row < N_NODES)
        z[(size_t)row * OUT_DIM + (nt << 4) + ncol] = acc[nt][r];
    }
  }
}

// ---------------------------------------------------------------------------
// Kernel 2: per-node attention scalars s1[n] = z[n]·w1, s2[n] = z[n]·w2.
// One wave per node: 32 lanes x float4 = 128 = OUT_DIM.
// ---------------------------------------------------------------------------
__launch_bounds__(256)
__global__ void gat_node_attn(const float* __restrict__ z,
                              const float* __restrict__ attn_w,
                              float* __restrict__ s1, float* __restrict__ s2) {
  const int wid  = (int)((blockIdx.x * blockDim.x + threadIdx.x) >> 5);
  const int lane = threadIdx.x & 31;
  if (wid >= N_NODES) return;
  const float4 zv = ((const float4*)(z + (size_t)wid * OUT_DIM))[lane];
  const float4 w1 = ((const float4*)attn_w)[lane];       // attn_w[0][0:128]
  const float4 w2 = ((const float4*)attn_w)[32 + lane];  // attn_w[0][128:256]
  float d1 = zv.x * w1.x + zv.y * w1.y + zv.z * w1.z + zv.w * w1.w;
  float d2 = zv.x * w2.x + zv.y * w2.y + zv.z * w2.z + zv.w * w2.w;
#pragma unroll
  for (int o = 16; o > 0; o >>= 1) {
    d1 += __shfl_xor(d1, o, 32);
    d2 += __shfl_xor(d2, o, 32);
  }
  if (lane == 0) { s1[wid] = d1; s2[wid] = d2; }
}

// ---------------------------------------------------------------------------
// Kernel 3: sum(dist) for mu = mean(dist).
// ---------------------------------------------------------------------------
__global__ void gat_dist_sum(const int* __restrict__ dist, float* __restrict__ sum) {
  const int i = (int)(blockIdx.x * blockDim.x + threadIdx.x);
  float v = (i < N_EDGES) ? (float)dist[i] : 0.f;
#pragma unroll
  for (int o = 16; o > 0; o >>= 1) v += __shfl_xor(v, o, 32);
  if ((threadIdx.x & 31) == 0) atomicAdd(sum, v);
}

// Float atomic max via signed-max / unsigned-min bit trick.
// Identity init: 0xFFFFFFFF works for both paths (acts like -inf).
__device__ __forceinline__ void atomicMaxFloat(float* addr, float val) {
  if (val >= 0.f) atomicMax((int*)addr, __float_as_int(val));
  else            atomicMin((unsigned int*)addr, __float_as_uint(val));
}

// ---------------------------------------------------------------------------
// Kernel 4: per-edge logit e = poisson_pmf * leaky_relu(s1[src]+s2[dst]),
// plus segment max into m[dst].
// ---------------------------------------------------------------------------
__global__ void gat_edge_e(const float* __restrict__ s1, const float* __restrict__ s2,
                           const int* __restrict__ src, const int* __restrict__ dst,
                           const int* __restrict__ dist, const float* __restrict__ sum,
                           float* __restrict__ e, float* __restrict__ m) {
  const int i = (int)(blockIdx.x * blockDim.x + threadIdx.x);
  if (i >= N_EDGES) return;
  const float a  = s1[src[i]] + s2[dst[i]];
  const float lr = (a >= 0.f) ? a : NEG_SLOPE * a;
  const float mu = *sum * (1.0f / (float)N_EDGES);
  const float k  = floorf((float)dist[i] * 0.5f);
  const float p  = expf(k * logf(mu) - mu - lgammaf(k + 1.0f));
  const float ev = p * lr;
  e[i] = ev;
  atomicMaxFloat(m + dst[i], ev);
}

// ---------------------------------------------------------------------------
// Kernel 5: ex = exp(e - m[dst]); denom[dst] += ex.
// ---------------------------------------------------------------------------
__global__ void gat_edge_exp(const float* __restrict__ e, const int* __restrict__ dst,
                             const float* __restrict__ m, float* __restrict__ ex,
                             float* __restrict__ denom) {
  const int i = (int)(blockIdx.x * blockDim.x + threadIdx.x);
  if (i >= N_EDGES) return;
  const int d   = dst[i];
  const float x = expf(e[i] - m[d]);
  ex[i] = x;
  atomicAdd(denom + d, x);
}

// ---------------------------------------------------------------------------
// Kernel 6: h_out[dst] += (ex/denom[dst]) * z[src].  One wave per edge,
// lane covers 4 floats (32 x 4 = 128). z is L2-resident (25.6 MB << 192 MB).
// ---------------------------------------------------------------------------
__launch_bounds__(256)
__global__ void gat_scatter(const int* __restrict__ src, const int* __restrict__ dst,
                            const float* __restrict__ ex, const float* __restrict__ denom,
                            const float* __restrict__ z, float* __restrict__ out) {
  const int edge = (int)((blockIdx.x * blockDim.x + threadIdx.x) >> 5);
  const int lane = threadIdx.x & 31;
  if (edge >= N_EDGES) return;  // wave-uniform exit
  const int d       = dst[edge];
  const float alpha = ex[edge] / denom[d];
  const float4 zv   = ((const float4*)(z + (size_t)src[edge] * OUT_DIM))[lane];
  float* o = out + (size_t)d * OUT_DIM + (lane << 2);
  atomicAdd(o + 0, alpha * zv.x);
  atomicAdd(o + 1, alpha * zv.y);
  atomicAdd(o + 2, alpha * zv.z);
  atomicAdd(o + 3, alpha * zv.w);
}

// ---------------------------------------------------------------------------
extern "C" void kernel_launch(void* const* d_in, const int* in_sizes, int n_in,
                              void* d_out, int out_size, void* d_ws, size_t ws_size,
                              hipStream_t stream) {
  const float* h    = (const float*)d_in[0];
  const float* fcw  = (const float*)d_in[1];
  const float* attn = (const float*)d_in[2];
  const int*   src  = (const int*)d_in[3];
  const int*   dst  = (const int*)d_in[4];
  const int*   dist = (const int*)d_in[5];
  float* out = (float*)d_out;

  // Workspace layout (floats): ~30.4 MB total.
  float* ws    = (float*)d_ws;
  float* z     = ws;                                   // 50000*128
  float* s1    = z + (size_t)N_NODES * OUT_DIM;        // 50000
  float* s2    = s1 + N_NODES;                         // 50000
  float* e     = s2 + N_NODES;                         // 500000
  float* ex    = e + N_EDGES;                          // 500000
  float* m     = ex + N_EDGES;                         // 50000
  float* denom = m + N_NODES;                          // 50000
  float* sum   = denom + N_NODES;                      // 1

  // Accumulator init (graph-capture-legal async memsets).
  (void)hipMemsetAsync(out, 0, (size_t)N_NODES * OUT_DIM * sizeof(float), stream);
  (void)hipMemsetAsync(denom, 0, (size_t)N_NODES * sizeof(float), stream);
  (void)hipMemsetAsync(sum, 0, sizeof(float), stream);
  (void)hipMemsetAsync(m, 0xFF, (size_t)N_NODES * sizeof(float), stream);  // -inf identity

  gat_gemm_z<<<(N_NODES + 127) / 128, 256, 0, stream>>>(h, fcw, z);
  gat_node_attn<<<(N_NODES + 7) / 8, 256, 0, stream>>>(z, attn, s1, s2);
  gat_dist_sum<<<(N_EDGES + 255) / 256, 256, 0, stream>>>(dist, sum);
  gat_edge_e<<<(N_EDGES + 255) / 256, 256, 0, stream>>>(s1, s2, src, dst, dist, sum, e, m);
  gat_edge_exp<<<(N_EDGES + 255) / 256, 256, 0, stream>>>(e, dst, m, ex, denom);
  gat_scatter<<<(N_EDGES + 7) / 8, 256, 0, stream>>>(src, dst, ex, denom, z, out);
}